// Encoder_15728170238401
// MI455X (gfx1250) — compile-verified
//
#include <hip/hip_runtime.h>
#include <math.h>

// ---------------- model constants (match reference) ----------------
#define NQ     10000      // HQ*WQ = 100*100
#define CDIM   256
#define NHEAD  8
#define HDD    32         // CDIM / NHEAD
#define NK     19840
#define NCAM   6
#define SSEQ   2          // temporal sequences
#define PT     4          // temporal points (LT=1)
#define LS     4          // spatial levels
#define ZA     4          // z anchors
#define PS     2          // spatial points

typedef float v2f __attribute__((ext_vector_type(2)));
typedef float v8f __attribute__((ext_vector_type(8)));

// =====================================================================
// fp32 WMMA GEMM, LDS-tiled:  C[M,N] = A[M,256] * W[256,N] + bias[N]
// Workgroup: 256 threads (8 waves) -> 128x64 output tile.
// Wave w: rows [w*16, w*16+16) x all 64 cols -> 4 accumulators.
// K-chunk = 32: A(128x32) + B(32x64, transposed) staged in LDS,
// 32 x V_WMMA_F32_16X16X4_F32 per wave per chunk from ds_load_b64 frags.
// N is a template constant (64/128/256/512) -> compile-time addressing.
// =====================================================================
template <int N>
__global__ __launch_bounds__(256) void gemm_wmma_tiled(
    const float* __restrict__ A, const float* __restrict__ W,
    const float* __restrict__ bias, float* __restrict__ C, int M) {
  constexpr int KC  = 32;
  constexpr int STR = 36;                 // padded row stride (floats), 16B-aligned
  __shared__ float lAs[128 * STR];        // A chunk [row][k]
  __shared__ float lBs[64 * STR];         // B chunk [n][k]  (transposed)

  const int m0   = blockIdx.x * 128;
  const int n0   = blockIdx.y * 64;
  const int tid  = threadIdx.x;           // 0..255
  const int wave = tid >> 5;              // 0..7
  const int lane = tid & 31;
  const int half = lane >> 4;             // K-pair select
  const int l15  = lane & 15;

  v8f acc[4] = {{}, {}, {}, {}};

  for (int k0 = 0; k0 < 256; k0 += KC) {
    // ---- stage A chunk: 128 rows x 32 k, coalesced float4 (b128) loads
    #pragma unroll
    for (int i = 0; i < 4; ++i) {
      int idx4 = tid + 256 * i;           // 1024 float4s total
      int row  = idx4 >> 3;               // 8 float4 per row
      int kq   = (idx4 & 7) * 4;
      float4 v = make_float4(0.f, 0.f, 0.f, 0.f);
      if (m0 + row < M)
        v = *(const float4*)(A + (size_t)(m0 + row) * 256 + k0 + kq);
      *(float4*)(&lAs[row * STR + kq]) = v;
    }
    // ---- stage B chunk: 32 k x 64 n, coalesced rows, store transposed [n][k]
    #pragma unroll
    for (int i = 0; i < 8; ++i) {
      int idx = tid + 256 * i;            // 2048 floats total
      int kk  = idx >> 6;
      int nn  = idx & 63;
      lBs[nn * STR + kk] = W[(size_t)(k0 + kk) * N + n0 + nn];
    }
    __syncthreads();

    // ---- 8 k-steps x 4 n-tiles = 32 WMMAs; A frag reused across 4 tiles
    const float* aRow = &lAs[(wave * 16 + l15) * STR + 2 * half];
    #pragma unroll
    for (int ks = 0; ks < KC; ks += 4) {
      v2f a = *(const v2f*)(aRow + ks);
      #pragma unroll
      for (int nt = 0; nt < 4; ++nt) {
        v2f b = *(const v2f*)(&lBs[(nt * 16 + l15) * STR + ks + 2 * half]);
        acc[nt] = __builtin_amdgcn_wmma_f32_16x16x4_f32(
            /*neg_a=*/false, a, /*neg_b=*/false, b,
            /*c_mod=*/(short)0, acc[nt], /*reuse_a=*/false, /*reuse_b=*/false);
      }
    }
    __syncthreads();
  }

  // ---- epilogue: bias + store (C/D layout: VGPR j -> row j + 8*half, col l15)
  #pragma unroll
  for (int nt = 0; nt < 4; ++nt) {
    float bv = bias ? bias[n0 + nt * 16 + l15] : 0.f;
    #pragma unroll
    for (int j = 0; j < 8; ++j) {
      int m = m0 + wave * 16 + j + 8 * half;
      if (m < M) C[(size_t)m * N + n0 + nt * 16 + l15] = acc[nt][j] + bv;
    }
  }
}

// =====================================================================
// q = q_origin @ projQ_w + projQ_b + q_pos        (K=3, tiny -> scalar)
// =====================================================================
__global__ void q_init_kernel(const float* __restrict__ qo,
                              const float* __restrict__ qpos,
                              const float* __restrict__ pw,
                              const float* __restrict__ pb,
                              float* __restrict__ q) {
  int i = blockIdx.x * blockDim.x + threadIdx.x;
  if (i >= NQ * CDIM) return;
  int c = i & 255, qi = i >> 8;
  float s = qo[qi * 3 + 0] * pw[c] + qo[qi * 3 + 1] * pw[256 + c] +
            qo[qi * 3 + 2] * pw[512 + c];
  q[i] = s + pb[c] + qpos[i];
}

// =====================================================================
// in-place softmax over contiguous groups of glen
// =====================================================================
__global__ void softmax_groups(float* __restrict__ x, int ngroups, int glen) {
  int g = blockIdx.x * blockDim.x + threadIdx.x;
  if (g >= ngroups) return;
  float* p = x + (size_t)g * glen;
  float mx = -1e30f;
  for (int i = 0; i < glen; ++i) mx = fmaxf(mx, p[i]);
  float s = 0.f;
  for (int i = 0; i < glen; ++i) { float e = expf(p[i] - mx); p[i] = e; s += e; }
  float inv = 1.f / s;
  for (int i = 0; i < glen; ++i) p[i] *= inv;
}

// =====================================================================
// bilinear sample, zero padding, align_corners=False (grid_sample semantics)
// =====================================================================
__device__ inline float bilin_zp(const float* __restrict__ v, int H, int W,
                                 int chan, float lx, float ly) {
  float x = lx * (float)W - 0.5f;
  float y = ly * (float)H - 0.5f;
  float x0 = floorf(x), y0 = floorf(y);
  float acc = 0.f;
  #pragma unroll
  for (int dy = 0; dy < 2; ++dy) {
    #pragma unroll
    for (int dx = 0; dx < 2; ++dx) {
      float xi = x0 + (float)dx, yi = y0 + (float)dy;
      if (xi >= 0.f && xi <= (float)(W - 1) && yi >= 0.f && yi <= (float)(H - 1)) {
        float w = (1.f - fabsf(x - xi)) * (1.f - fabsf(y - yi));
        acc += w * v[((int)yi * W + (int)xi) * CDIM + chan];
      }
    }
  }
  return acc;
}

// =====================================================================
// temporal MS-deformable attention (grid 100x100), result pre-scaled 1/S
// =====================================================================
__global__ void temp_deform(const float* __restrict__ vt,
                            const float* __restrict__ toff,
                            const float* __restrict__ tattn,
                            float* __restrict__ tout) {
  int i = blockIdx.x * blockDim.x + threadIdx.x;
  if (i >= NQ * CDIM) return;
  int d = i & 31, h = (i >> 5) & 7, qi = i >> 8;
  float refx = ((float)(qi % 100) + 0.5f) * 0.01f;
  float refy = ((float)(qi / 100) + 0.5f) * 0.01f;
  int chan = h * HDD + d;
  float acc = 0.f;
  for (int s = 0; s < SSEQ; ++s) {
    const float* offp = toff + (size_t)qi * 128 + (s * NHEAD + h) * (PT * 2);
    const float* wp   = tattn + (size_t)qi * 64 + (s * NHEAD + h) * PT;
    #pragma unroll
    for (int p = 0; p < PT; ++p) {
      float lx = refx + offp[p * 2 + 0] * 0.01f;   // / WQ
      float ly = refy + offp[p * 2 + 1] * 0.01f;   // / HQ
      acc += wp[p] * bilin_zp(vt, 100, 100, chan, lx, ly);
    }
  }
  tout[i] = acc * 0.5f;   // 1/S
}

// =====================================================================
// out = LayerNorm(X + R) * g + b      (wave32 per row, 8 elems/lane)
// =====================================================================
__global__ __launch_bounds__(256) void add_ln(const float* __restrict__ X,
                                              const float* __restrict__ R,
                                              const float* __restrict__ g,
                                              const float* __restrict__ b,
                                              float* __restrict__ out) {
  int wave = threadIdx.x >> 5, lane = threadIdx.x & 31;
  int row = blockIdx.x * 8 + wave;
  if (row >= NQ) return;
  const float* xr = X + (size_t)row * CDIM;
  const float* rr = R + (size_t)row * CDIM;
  float vals[8];
  float sum = 0.f;
  #pragma unroll
  for (int j = 0; j < 8; ++j) {
    float t = xr[lane + j * 32] + rr[lane + j * 32];
    vals[j] = t; sum += t;
  }
  #pragma unroll
  for (int off = 16; off > 0; off >>= 1) sum += __shfl_xor(sum, off, 32);
  float mean = sum * (1.f / 256.f);
  float var = 0.f;
  #pragma unroll
  for (int j = 0; j < 8; ++j) { float t = vals[j] - mean; var += t * t; }
  #pragma unroll
  for (int off = 16; off > 0; off >>= 1) var += __shfl_xor(var, off, 32);
  var *= (1.f / 256.f);
  float inv = rsqrtf(var + 1e-5f);
  float* o = out + (size_t)row * CDIM;
  #pragma unroll
  for (int j = 0; j < 8; ++j) {
    int c = lane + j * 32;
    o[c] = (vals[j] - mean) * inv * g[c] + b[c];
  }
}

__global__ void fill_zero(float* __restrict__ p, int n) {
  int i = blockIdx.x * blockDim.x + threadIdx.x;
  if (i < n) p[i] = 0.f;
}

// =====================================================================
// spatial MS-deformable attention for one camera, accumulated into slots
// =====================================================================
__global__ void spat_deform(const float* __restrict__ sval,
                            const float* __restrict__ soff,
                            const float* __restrict__ sattn,
                            const float* __restrict__ refc,   // [NQ,ZA,2] for this cam
                            const unsigned char* __restrict__ mask, // [NQ,ZA]
                            float* __restrict__ slots) {
  int i = blockIdx.x * blockDim.x + threadIdx.x;
  if (i >= NQ * CDIM) return;
  int d = i & 31, h = (i >> 5) & 7, qi = i >> 8;
  const unsigned char* m = mask + (size_t)qi * ZA;
  if (!(m[0] | m[1] | m[2] | m[3])) return;   // mask_any == 0 -> contributes nothing

  const int lvlH[4] = {93, 47, 24, 12};
  const int lvlW[4] = {160, 80, 40, 20};
  const int lvlS[4] = {0, 14880, 18640, 19600};

  int chan = h * HDD + d;
  const float* offp = soff  + (size_t)qi * 512 + h * (LS * ZA * PS * 2);
  const float* wp   = sattn + (size_t)qi * 256 + h * (LS * ZA * PS);
  const float* refp = refc  + (size_t)qi * (ZA * 2);

  float acc = 0.f;
  for (int l = 0; l < LS; ++l) {
    int H = lvlH[l], W = lvlW[l];
    const float* vlvl = sval + (size_t)lvlS[l] * CDIM;
    float invW = 1.f / (float)W, invH = 1.f / (float)H;
    #pragma unroll
    for (int z = 0; z < ZA; ++z) {
      float rx = refp[z * 2 + 0], ry = refp[z * 2 + 1];
      #pragma unroll
      for (int ps = 0; ps < PS; ++ps) {
        int o = ((l * ZA + z) * PS + ps) * 2;
        float lx = rx + offp[o + 0] * invW;
        float ly = ry + offp[o + 1] * invH;
        acc += wp[l * (ZA * PS) + z * PS + ps] * bilin_zp(vlvl, H, W, chan, lx, ly);
      }
    }
  }
  slots[i] += acc;
}

// slots[qi,:] /= max(sum_{cam,z} mask, 1)
__global__ void count_scale(const unsigned char* __restrict__ mask,
                            float* __restrict__ slots) {
  int i = blockIdx.x * blockDim.x + threadIdx.x;
  if (i >= NQ * CDIM) return;
  int qi = i >> 8;
  int cnt = 0;
  for (int c = 0; c < NCAM; ++c) {
    const unsigned char* m = mask + ((size_t)c * NQ + qi) * ZA;
    cnt += (int)m[0] + (int)m[1] + (int)m[2] + (int)m[3];
  }
  slots[i] *= 1.f / fmaxf((float)cnt, 1.f);
}

// =====================================================================
extern "C" void kernel_launch(void* const* d_in, const int* in_sizes, int n_in,
                              void* d_out, int out_size, void* d_ws, size_t ws_size,
                              hipStream_t stream) {
  (void)in_sizes; (void)n_in; (void)out_size; (void)ws_size;
  // ---- inputs (setup_inputs dict order; params flattened in insertion order)
  const float*         spat_value = (const float*)d_in[1];
  const float*         ref_cam    = (const float*)d_in[3];   // [6,1,NQ,ZA,2]
  const unsigned char* bev_mask   = (const unsigned char*)d_in[4]; // [6,1,NQ,ZA] bool
  const float* q_origin  = (const float*)d_in[6];
  const float* q_pos     = (const float*)d_in[7];
  const float* projQ_w   = (const float*)d_in[8];
  const float* projQ_b   = (const float*)d_in[9];
  const float* t_value_w = (const float*)d_in[10];
  const float* t_value_b = (const float*)d_in[11];
  const float* t_off_w   = (const float*)d_in[12];
  const float* t_off_b   = (const float*)d_in[13];
  const float* t_attn_w  = (const float*)d_in[14];
  const float* t_attn_b  = (const float*)d_in[15];
  const float* t_out_w   = (const float*)d_in[16];
  const float* t_out_b   = (const float*)d_in[17];
  const float* s_value_w = (const float*)d_in[18];
  const float* s_value_b = (const float*)d_in[19];
  const float* s_off_w   = (const float*)d_in[20];
  const float* s_off_b   = (const float*)d_in[21];
  const float* s_attn_w  = (const float*)d_in[22];
  const float* s_attn_b  = (const float*)d_in[23];
  const float* s_out_w   = (const float*)d_in[24];
  const float* s_out_b   = (const float*)d_in[25];
  const float* ln1_g = (const float*)d_in[26];
  const float* ln1_b = (const float*)d_in[27];
  const float* ln2_g = (const float*)d_in[28];
  const float* ln2_b = (const float*)d_in[29];
  const float* ln3_g = (const float*)d_in[30];
  const float* ln3_b = (const float*)d_in[31];
  const float* ffn_w = (const float*)d_in[32];
  const float* ffn_b = (const float*)d_in[33];

  float* ws = (float*)d_ws;
  const size_t M1 = (size_t)NQ * CDIM;           // 2,560,000
  // persistent
  float* a1    = ws;                              // [NQ,256]
  // phase 1 (temporal) region starting at R = M1
  float* q     = ws + M1;                         // [NQ,256]
  float* vt    = ws + M1 * 2;                     // [NQ,256]
  float* toff  = ws + M1 * 3;                     // [NQ,128]
  float* tattn = ws + M1 * 3 + 1280000;           // [NQ,64]
  float* tout  = ws + M1 * 3 + 1920000;           // [NQ,256]
  float* tproj = ws + M1 * 4 + 1920000;           // [NQ,256]
  // phase 3 (spatial) region reuses from R
  float* soff  = ws + M1;                         // [NQ,512]
  float* sattn = ws + M1 * 3;                     // [NQ,256]
  float* sval  = ws + M1 * 4;                     // [NK,256] (per-cam staging)
  float* slots = ws + M1 * 4 + (size_t)NK * CDIM; // [NQ,256]
  // phase 4 reuses soff/sattn regions
  float* sout   = ws + M1;                        // [NQ,256]
  float* a2     = ws + M1 * 2;                    // [NQ,256]
  float* ffnbuf = ws + M1 * 3;                    // [NQ,256]

  dim3 blk256(256);
  const int gQC = (NQ * CDIM + 255) / 256;

  auto gemm = [&](const float* A, const float* Wm, const float* bias, float* Cc,
                  int M, int N) {
    dim3 grid((M + 127) / 128, N / 64), block(256);
    switch (N) {
      case 64:
        gemm_wmma_tiled<64><<<grid, block, 0, stream>>>(A, Wm, bias, Cc, M);
        break;
      case 128:
        gemm_wmma_tiled<128><<<grid, block, 0, stream>>>(A, Wm, bias, Cc, M);
        break;
      case 256:
        gemm_wmma_tiled<256><<<grid, block, 0, stream>>>(A, Wm, bias, Cc, M);
        break;
      case 512:
        gemm_wmma_tiled<512><<<grid, block, 0, stream>>>(A, Wm, bias, Cc, M);
        break;
    }
  };

  // ---- learned BEV query
  q_init_kernel<<<gQC, blk256, 0, stream>>>(q_origin, q_pos, projQ_w, projQ_b, q);

  // ---- temporal self-attention
  gemm(q, t_value_w, t_value_b, vt,   NQ, 256);
  gemm(q, t_off_w,   t_off_b,   toff, NQ, 128);
  gemm(q, t_attn_w,  t_attn_b,  tattn, NQ, 64);
  softmax_groups<<<(NQ * SSEQ * NHEAD + 255) / 256, blk256, 0, stream>>>(
      tattn, NQ * SSEQ * NHEAD, PT);
  temp_deform<<<gQC, blk256, 0, stream>>>(vt, toff, tattn, tout);
  gemm(tout, t_out_w, t_out_b, tproj, NQ, 256);
  add_ln<<<NQ / 8, blk256, 0, stream>>>(tproj, q, ln1_g, ln1_b, a1);

  // ---- spatial cross-attention
  gemm(a1, s_off_w,  s_off_b,  soff,  NQ, 512);
  gemm(a1, s_attn_w, s_attn_b, sattn, NQ, 256);
  softmax_groups<<<(NQ * NHEAD + 255) / 256, blk256, 0, stream>>>(
      sattn, NQ * NHEAD, LS * ZA * PS);
  fill_zero<<<gQC, blk256, 0, stream>>>(slots, NQ * CDIM);
  for (int c = 0; c < NCAM; ++c) {
    gemm(spat_value + (size_t)c * NK * CDIM, s_value_w, s_value_b, sval, NK, 256);
    spat_deform<<<gQC, blk256, 0, stream>>>(
        sval, soff, sattn,
        ref_cam + (size_t)c * NQ * ZA * 2,
        bev_mask + (size_t)c * NQ * ZA, slots);
  }
  count_scale<<<gQC, blk256, 0, stream>>>(bev_mask, slots);
  gemm(slots, s_out_w, s_out_b, sout, NQ, 256);
  add_ln<<<NQ / 8, blk256, 0, stream>>>(sout, a1, ln2_g, ln2_b, a2);

  // ---- FFN (linear) + final LN
  gemm(a2, ffn_w, ffn_b, ffnbuf, NQ, 256);
  add_ln<<<NQ / 8, blk256, 0, stream>>>(ffnbuf, a2, ln3_g, ln3_b, (float*)d_out);
}